// RoiPoolingConv_11957188952220
// MI455X (gfx1250) — compile-verified
//
#include <hip/hip_runtime.h>
#include <hip/hip_bf16.h>

// 3-D ROI max-pooling for MI455X (gfx1250, wave32).
// img : (1, 48, 48, 48, 128) f32   rois : (1, 64, 4) f32 (cx,cy,cz,r)
// out : (1, 64, 7, 7, 7, 128) f32  out[roi,k,j,i,c] pools axis0-bin i,
// axis1-bin j, axis2-bin k (reference transposes (2,1,0,3)).
//
// Memory-bound gather. img (56.6 MB) is kept L2-resident (192 MB L2, RT
// loads); output (11.2 MB) is streamed once with non-temporal stores so it
// does not evict img. One wave32 x float4 covers a full 512 B channel line
// per voxel (coalesced global_load_b128). 8 waves/block hide L2 latency.
// No matmul structure -> WMMA/SWMMAC not applicable.

static constexpr int Dn = 48;      // spatial extent per axis
static constexpr int Pn = 7;       // pool size
static constexpr int Cn = 128;     // channels
static constexpr int C4 = Cn / 4;  // float4 quads per voxel (32 == wave32)
static constexpr int NR = 64;      // rois
static constexpr int NB = Pn * Pn; // 49 xy-bins per (roi, z-bin) block

// Native clang vector: required by __builtin_nontemporal_store (HIP's float4
// class type is rejected); lowers to the same b128 load/store.
typedef __attribute__((ext_vector_type(4))) float v4f;

__global__ __launch_bounds__(256, 4)
void roi_maxpool3d_kernel(const float* __restrict__ img,
                          const float* __restrict__ rois,
                          float* __restrict__ out)
{
    __shared__ int  s_mn[3][Pn];
    __shared__ int  s_ext[3][Pn];
    __shared__ int4 s_bin[NB];     // (mn1, e1, mn2, e2) per flat xy-bin

    const int blk = blockIdx.x;
    const int roi = blk / Pn;
    const int k   = blk - roi * Pn;      // z-bin (axis2)
    const int tid = threadIdx.x;

    // ---- per-ROI axis bounds, faithful to the reference float semantics ----
    if (tid < 3 * Pn) {
        const int axis = tid / Pn;
        const int bin  = tid - axis * Pn;
        const float c  = rois[roi * 4 + axis];
        const float r  = rois[roi * 4 + 3];
        const float rl2 = (2.0f * r) / (float)Pn;
        const float mnf = (c - r) + (float)bin * rl2;    // mnf >= 0 always
        int mn = (int)mnf;                                // truncating cast
        int mx = (int)(mnf + rl2);
        mn = min(max(mn, 1), Dn - 2);
        int ext = mx - mn;                                // uses clamped mn
        if (ext < 2) ext = 2;
        mx = min(Dn, mn + ext);
        s_mn[axis][bin]  = mn;
        s_ext[axis][bin] = mx - mn;                       // always in [2, 5]
    }
    __syncthreads();

    // ---- flatten to a 49-entry bin table: kills div/mod-by-7 in hot loop ----
    if (tid < NB) {
        const int i = tid % Pn;           // x-bin (axis0)
        const int j = tid / Pn;           // y-bin (axis1)
        s_bin[tid] = make_int4(s_mn[0][i], s_ext[0][i], s_mn[1][j], s_ext[1][j]);
    }
    __syncthreads();

    const int c4  = tid & (C4 - 1);   // channel quad 0..31 (one wave = 128 ch)
    const int sub = tid >> 5;         // wave id 0..7 -> bin subgroup

    const int mn3 = s_mn[2][k];
    const int e3  = s_ext[2][k];

    const v4f* __restrict__ img4 = (const v4f*)img;
    v4f*       __restrict__ out4 = (v4f*)out;

    // out[roi,k,j,i,c]: (j*7+i) == b, so offset collapses to a linear form.
    const size_t obase = ((size_t)roi * Pn + (size_t)k) * NB;

    for (int b = sub; b < NB; b += 8) {
        const int4 bb  = s_bin[b];
        const int  mn1 = bb.x, e1 = bb.y;
        const int  mn2 = bb.z, e2 = bb.w;

        // Warm L2 for this wave's next bin (global_prefetch_b8).
        const int bn = b + 8;
        if (bn < NB) {
            const int4 nb = s_bin[bn];
            const size_t pa =
                ((size_t)(nb.x * Dn + nb.z) * Dn + (size_t)mn3) * Cn;
            __builtin_prefetch((const void*)(img + pa), 0, 1);
        }

        v4f acc;
        acc.x = acc.y = acc.z = acc.w = -__builtin_inff();

        for (int x = 0; x < e1; ++x) {
            const size_t rowx = (size_t)(mn1 + x) * Dn;
            for (int y = 0; y < e2; ++y) {
                const size_t base =
                    ((rowx + (size_t)(mn2 + y)) * Dn + (size_t)mn3) * C4 + c4;
                #pragma unroll 2
                for (int z = 0; z < e3; ++z) {
                    // 32 lanes x 16 B = full 512 B channel line per voxel.
                    const v4f v = img4[base + (size_t)z * C4];
                    acc.x = fmaxf(acc.x, v.x);
                    acc.y = fmaxf(acc.y, v.y);
                    acc.z = fmaxf(acc.z, v.z);
                    acc.w = fmaxf(acc.w, v.w);
                }
            }
        }

        // Output is write-once/never-re-read: non-temporal store keeps the
        // L2 free for the (reused) img volume.
        __builtin_nontemporal_store(acc, out4 + (obase + (size_t)b) * C4 + c4);
    }
}

extern "C" void kernel_launch(void* const* d_in, const int* in_sizes, int n_in,
                              void* d_out, int out_size, void* d_ws, size_t ws_size,
                              hipStream_t stream) {
    (void)in_sizes; (void)n_in; (void)d_ws; (void)ws_size; (void)out_size;
    const float* img  = (const float*)d_in[0];   // (1,48,48,48,128) f32
    const float* rois = (const float*)d_in[1];   // (1,64,4) f32
    float* out = (float*)d_out;                  // (1,64,7,7,7,128) f32

    dim3 grid(NR * Pn);   // 448 workgroups: one per (roi, z-bin)
    dim3 block(256);      // 8 wave32 waves
    hipLaunchKernelGGL(roi_maxpool3d_kernel, grid, block, 0, stream,
                       img, rois, out);
}